// RPEMultiHeadAttention_71829033058397
// MI455X (gfx1250) — compile-verified
//
#include <hip/hip_runtime.h>
#include <cstddef>
#include <cstdint>

// RPE Multi-Head Attention, fused for MI455X (gfx1250).
// b=2, n=m=512, d=256, h=8, c=32.
// scores_p = E . u + qb fusion avoids the 68.7-GFLOP p-projection; the kernel
// is HBM-bound on the single 536MB read of embed_qk (~23us floor at 23.3TB/s).
// E is streamed via the Tensor Data Mover (tensor_load_to_lds, TENSORcnt) into
// double-buffered LDS so DMA overlaps the f32 WMMA chains.

#define SCALE 0.17677669529663687f // 1/sqrt(32)

typedef float v2f __attribute__((ext_vector_type(2)));
typedef float v8f __attribute__((ext_vector_type(8)));
typedef unsigned int u32x4 __attribute__((ext_vector_type(4)));
typedef int i32x4 __attribute__((ext_vector_type(4)));
typedef int i32x8 __attribute__((ext_vector_type(8)));

static __device__ __forceinline__ int lane_id() { return threadIdx.x & 31; }

// Fragment where the K-index is contiguous in memory (A row-major MxK, or
// B supplied as N-major rows with K contiguous). Per ISA 16x16x4 layout:
// VGPR0 = K=k0 (lanes 0-15) / K=k0+2 (lanes 16-31); VGPR1 = K+1 / K+3.
static __device__ __forceinline__ v2f ld_fragK(const float* base, int ld) {
  int l = lane_id();
  const float* p = base + (l & 15) * ld + ((l >> 4) << 1);
  return *(const v2f*)p;
}

// B fragment from a row-major KxN matrix (e.g. weight matrices).
static __device__ __forceinline__ v2f ld_fragB_KN(const float* base, int ldn) {
  int l = lane_id();
  const float* p = base + (l & 15) + (((l >> 4) << 1) * ldn);
  v2f r;
  r.x = p[0];
  r.y = p[ldn];
  return r;
}

static __device__ __forceinline__ v8f wmma4(v2f a, v2f b, v8f c) {
  return __builtin_amdgcn_wmma_f32_16x16x4_f32(false, a, false, b, (short)0, c,
                                               false, false);
}

// ---------------------------------------------------------------------------
// TDM: issue a 2D tile load (128 rows x 256 f32, row stride 256 f32) from
// global memory into LDS at lds_byte_off, with LDS padding of 4 DWORDs every
// 256 DWORDs (-> 260-float padded row stride, conflict-free fragment reads).
// D# fields per CDNA5 ISA ch.8. Tracked by TENSORcnt. EXEC ignored.
// Toolchain uses the 6-arg (clang-23/therock) builtin form.
// ---------------------------------------------------------------------------
static __device__ __forceinline__ void tdm_load_tile128(const float* gsrc,
                                                        uint32_t lds_byte_off) {
  uint64_t ga = (uint64_t)(uintptr_t)gsrc;
  u32x4 g0;
  g0[0] = 1u;                  // count=1 (valid), is_restore=0, gather_mode=0
  g0[1] = lds_byte_off;        // lds_addr
  g0[2] = (uint32_t)ga;        // global_addr[31:0]
  g0[3] = (uint32_t)((ga >> 32) & 0x01FFFFFFu) | (2u << 30); // addr[56:32], type=2
  i32x8 g1;
  // data_size=2 (4B), pad_enable=1, pad_interval=7 (256 dw), pad_amount=3 (4 dw)
  g1[0] = (int)((2u << 16) | (1u << 20) | (7u << 22) | (3u << 25));
  g1[1] = (int)(256u << 16); // tensor_dim0[15:0]=256 at bits[63:48]
  g1[2] = (int)(512u << 16); // tensor_dim0[31:16]=0; tensor_dim1[15:0]=512
  g1[3] = (int)(256u << 16); // tensor_dim1[31:16]=0; tile_dim0=256
  g1[4] = (int)(128u);       // tile_dim1=128, tile_dim2=0
  g1[5] = (int)(256u);       // tensor_dim0_stride[31:0]=256
  g1[6] = 0;                 // stride0[47:32]=0, tensor_dim1_stride lo=0
  g1[7] = 0;
  i32x4 gz4 = {0, 0, 0, 0};          // groups 2/3 unused (2D tensor)
  i32x8 gz8 = {0, 0, 0, 0, 0, 0, 0, 0};
  __builtin_amdgcn_tensor_load_to_lds(g0, g1, gz4, gz4, gz8, 0);
}

// ---------------------------------------------------------------------------
// Kernel 1: C[M x N] = A[M x K] @ W[K x N] + bias  (QKV projections)
// ---------------------------------------------------------------------------
__global__ __launch_bounds__(256) void k_gemm_bias(
    const float* __restrict__ A, const float* __restrict__ W,
    const float* __restrict__ bias, float* __restrict__ Cout, int Mrows,
    int Ncols, int Kdim) {
  int wave = threadIdx.x >> 5;
  int tiles_n = Ncols >> 4;
  int tile = blockIdx.x * (blockDim.x >> 5) + wave;
  int total = (Mrows >> 4) * tiles_n;
  if (tile >= total) return; // wave-uniform exit; EXEC all-ones for WMMA
  int tm = tile / tiles_n, tn = tile % tiles_n;

  v8f acc = {0.f, 0.f, 0.f, 0.f, 0.f, 0.f, 0.f, 0.f};
  const float* Abase = A + (size_t)(tm * 16) * Kdim;
  const float* Wbase = W + tn * 16;
  for (int k0 = 0; k0 < Kdim; k0 += 4) {
    v2f a = ld_fragK(Abase + k0, Kdim);
    v2f b = ld_fragB_KN(Wbase + (size_t)k0 * Ncols, Ncols);
    acc = wmma4(a, b, acc);
  }
  int l = lane_id(), r = l & 15, hi = l >> 4;
  int col = tn * 16 + r;
  float bi = bias[col];
#pragma unroll
  for (int j = 0; j < 8; ++j) {
    int row = tm * 16 + j + 8 * hi;
    Cout[(size_t)row * Ncols + col] = acc[j] + bi;
  }
}

// ---------------------------------------------------------------------------
// Kernel 2: u[b,n,h,d0] = sum_c Wp[d0, h*32+c] * Q[b,n,h*32+c]
//           qb[b,n,h]   = sum_c bp[h*32+c]    * Q[b,n,h*32+c]
// ---------------------------------------------------------------------------
__global__ __launch_bounds__(256) void k_uproj(const float* __restrict__ Q,
                                               const float* __restrict__ Wp,
                                               const float* __restrict__ bp,
                                               float* __restrict__ U,
                                               float* __restrict__ qb) {
  int bn = blockIdx.x;
  int t = threadIdx.x;
  __shared__ float qrow[256];
  qrow[t] = Q[(size_t)bn * 256 + t];
  __syncthreads();
  const float* wrow = Wp + (size_t)t * 256; // row d0 = t
#pragma unroll
  for (int h = 0; h < 8; ++h) {
    float s = 0.f;
    const float* wh = wrow + h * 32;
    const float* qh = qrow + h * 32;
#pragma unroll 8
    for (int cc = 0; cc < 32; ++cc) s += wh[cc] * qh[cc];
    U[((size_t)bn * 8 + h) * 256 + t] = s;
  }
  if (t < 8) {
    float s = 0.f;
    for (int cc = 0; cc < 32; ++cc) s += bp[t * 32 + cc] * qrow[t * 32 + cc];
    qb[(size_t)bn * 8 + t] = s;
  }
}

// ---------------------------------------------------------------------------
// Kernel 3: Se[b,h,n,m] = sum_c Q[b,n,h*32+c] * Kp[b,m,h*32+c] (WMMA, K=32)
// ---------------------------------------------------------------------------
__global__ __launch_bounds__(256) void k_scores_e(const float* __restrict__ Q,
                                                  const float* __restrict__ Kp,
                                                  float* __restrict__ Se) {
  int wave = threadIdx.x >> 5;
  int tile = blockIdx.x * 8 + wave;
  int bh = tile >> 10;
  int tt = tile & 1023;
  int tm = tt >> 5, tn = tt & 31;
  int b = bh >> 3, h = bh & 7;
  const float* Ab = Q + ((size_t)b * 512 + tm * 16) * 256 + h * 32;
  const float* Bb = Kp + ((size_t)b * 512 + tn * 16) * 256 + h * 32;
  v8f acc = {0.f, 0.f, 0.f, 0.f, 0.f, 0.f, 0.f, 0.f};
#pragma unroll
  for (int k0 = 0; k0 < 32; k0 += 4)
    acc = wmma4(ld_fragK(Ab + k0, 256), ld_fragK(Bb + k0, 256), acc);
  int l = lane_id(), r = l & 15, hi = l >> 4;
#pragma unroll
  for (int j = 0; j < 8; ++j) {
    size_t row = ((size_t)(b * 8 + h) * 512 + tm * 16 + j + 8 * hi);
    Se[row * 512 + tn * 16 + r] = acc[j];
  }
}

// ---------------------------------------------------------------------------
// Main kernel: one block per (b,n). Memory-bound on embed_qk (512KB/block).
// TDM double-buffers 128x256 E tiles into LDS; 8 waves run f32 WMMA chains;
// then per-head softmax + attn@V.
// ---------------------------------------------------------------------------
__global__ __launch_bounds__(256) void k_attn_main(
    const float* __restrict__ E, const float* __restrict__ U,
    const float* __restrict__ qb, const float* __restrict__ Se,
    const float* __restrict__ Vp, float* __restrict__ hidden,
    float* __restrict__ attn) {
  constexpr int LDE = 260; // 256 + TDM pad (4 dw / 256 dw) -> conflict-free
  __shared__ float lE[2][128 * LDE]; // 266,240 B (double buffered)
  __shared__ float lU[16 * LDE];     //  16,640 B
  __shared__ float lS[8 * 512];      //  16,384 B
  __shared__ float lqb[8];           // total ~299.3 KB < 320 KB/WGP

  int bn = blockIdx.x;
  int b = bn >> 9, nn = bn & 511;
  int t = threadIdx.x;
  int wave = t >> 5, l = t & 31, r = l & 15, hi = l >> 4;

  const float* Ebn = E + (size_t)bn * 512 * 256;

  // Prologue: wave 0 kicks TDM loads of super-tiles 0 and 1.
  if (wave == 0) {
    if (l == 0) {
      tdm_load_tile128(Ebn, (uint32_t)(uintptr_t)&lE[0][0]);
      tdm_load_tile128(Ebn + 128 * 256, (uint32_t)(uintptr_t)&lE[1][0]);
    }
  }

  // Load U into LDS, zero-padding heads 8..15 (WMMA N=16).
  for (int i = t; i < 16 * 256; i += 256) {
    int h = i >> 8, dd = i & 255;
    lU[h * LDE + dd] = (h < 8) ? U[((size_t)bn * 8 + h) * 256 + dd] : 0.f;
  }
  if (t < 8) lqb[t] = qb[(size_t)bn * 8 + t];

  for (int st = 0; st < 4; ++st) {
    int cur = st & 1;
    // TDM completes in-order per wave: tensorcnt<=1 (or 0 at the end) means
    // tile `st` has fully landed in LDS. (Immediate must be a literal.)
    if (wave == 0) {
      if (st < 3) {
        __builtin_amdgcn_s_wait_tensorcnt(1);
      } else {
        __builtin_amdgcn_s_wait_tensorcnt(0);
      }
    }
    __syncthreads(); // release: buffer `cur` valid (also covers lU on st==0)

    v8f acc = {0.f, 0.f, 0.f, 0.f, 0.f, 0.f, 0.f, 0.f};
    const float* Ab = &lE[cur][wave * 16 * LDE];
#pragma unroll 4
    for (int k0 = 0; k0 < 256; k0 += 4) {
      v2f a = ld_fragK(Ab + k0, LDE);
      v2f bb = ld_fragK(&lU[k0], LDE); // B[k][head]: head = lane&15
      acc = wmma4(a, bb, acc);
    }
    int m0 = st * 128 + wave * 16;
    if (r < 8) { // only heads 0..7 are real
#pragma unroll
      for (int j = 0; j < 8; ++j) lS[r * 512 + m0 + j + 8 * hi] = acc[j];
    }
    __syncthreads(); // all waves done reading buffer `cur`
    if (st < 2 && wave == 0 && l == 0)
      tdm_load_tile128(Ebn + (size_t)(st + 2) * 128 * 256,
                       (uint32_t)(uintptr_t)&lE[cur][0]);
  }
  __syncthreads();

  // Softmax: wave w owns head w; 512 columns -> 16 per lane.
  {
    int h = wave;
    const float* serow = Se + ((size_t)(b * 8 + h) * 512 + nn) * 512;
    float qbh = lqb[h];
    float vals[16];
    float mx = -3.0e38f;
#pragma unroll
    for (int i = 0; i < 16; ++i) {
      int m = i * 32 + l;
      float s = (lS[h * 512 + m] + serow[m] + qbh) * SCALE;
      vals[i] = s;
      mx = fmaxf(mx, s);
    }
#pragma unroll
    for (int off = 16; off > 0; off >>= 1)
      mx = fmaxf(mx, __shfl_xor(mx, off, 32));
    float sum = 0.f;
#pragma unroll
    for (int i = 0; i < 16; ++i) {
      float e = __expf(vals[i] - mx);
      vals[i] = e;
      sum += e;
    }
#pragma unroll
    for (int off = 16; off > 0; off >>= 1) sum += __shfl_xor(sum, off, 32);
    float inv = 1.f / sum;
    float* arow = attn + ((size_t)(b * 8 + h) * 512 + nn) * 512;
#pragma unroll
    for (int i = 0; i < 16; ++i) {
      int m = i * 32 + l;
      float a = vals[i] * inv;
      lS[h * 512 + m] = a;
      arow[m] = a;
    }
  }
  __syncthreads();

  // hidden[b,n,h*32+c] = sum_m attn[h,m] * Vp[b,m,h*32+c]
  {
    int h = t >> 5, cix = t & 31;
    const float* Vb = Vp + (size_t)b * 512 * 256 + h * 32 + cix;
    float acc2 = 0.f;
#pragma unroll 4
    for (int m = 0; m < 512; ++m) acc2 += lS[h * 512 + m] * Vb[(size_t)m * 256];
    hidden[((size_t)(b * 512 + nn)) * 256 + h * 32 + cix] = acc2;
  }
}

// ---------------------------------------------------------------------------
extern "C" void kernel_launch(void* const* d_in, const int* in_sizes, int n_in,
                              void* d_out, int out_size, void* d_ws,
                              size_t ws_size, hipStream_t stream) {
  (void)in_sizes;
  (void)n_in;
  (void)out_size;
  (void)ws_size;
  const float* input_q = (const float*)d_in[0];
  const float* input_k = (const float*)d_in[1];
  const float* input_v = (const float*)d_in[2];
  const float* embed_qk = (const float*)d_in[3];
  const float* Wq = (const float*)d_in[4];
  const float* bq = (const float*)d_in[5];
  const float* Wk = (const float*)d_in[6];
  const float* bk = (const float*)d_in[7];
  const float* Wv = (const float*)d_in[8];
  const float* bv = (const float*)d_in[9];
  const float* Wp = (const float*)d_in[10];
  const float* bp = (const float*)d_in[11];

  float* ws = (float*)d_ws;
  float* Q = ws;                 //   262,144 f
  float* Kp = Q + 262144;        //   262,144 f
  float* Vp = Kp + 262144;       //   262,144 f
  float* U = Vp + 262144;        // 2,097,152 f  (b,n,8,256)
  float* qb = U + 2097152;       //     8,192 f
  float* Se = qb + 8192;         // 4,194,304 f  (b,h,n,m)  -> ~28.3 MB total

  float* hidden = (float*)d_out;        // (b,n,256) = 262,144 f
  float* attn = hidden + 2 * 512 * 256; // (b,h,n,m) = 4,194,304 f

  k_gemm_bias<<<128, 256, 0, stream>>>(input_q, Wq, bq, Q, 1024, 256, 256);
  k_gemm_bias<<<128, 256, 0, stream>>>(input_k, Wk, bk, Kp, 1024, 256, 256);
  k_gemm_bias<<<128, 256, 0, stream>>>(input_v, Wv, bv, Vp, 1024, 256, 256);
  k_uproj<<<1024, 256, 0, stream>>>(Q, Wp, bp, U, qb);
  k_scores_e<<<2048, 256, 0, stream>>>(Q, Kp, Se);
  k_attn_main<<<1024, 256, 0, stream>>>(embed_qk, U, qb, Se, Vp, hidden, attn);
}